// GCN_85864986181814
// MI455X (gfx1250) — compile-verified
//
#include <hip/hip_runtime.h>
#include <stdint.h>

typedef __attribute__((ext_vector_type(2))) float v2f;
typedef __attribute__((ext_vector_type(4))) float v4f;
typedef __attribute__((ext_vector_type(8))) float v8f;

// ---------------------------------------------------------------------------
// Zero-fill (d_out holds the segment-sum accumulators; it is poisoned 0xAA).
// ---------------------------------------------------------------------------
__global__ void gcn_zero_f4(float* __restrict__ p, int n4) {
    int i = blockIdx.x * blockDim.x + threadIdx.x;
    if (i < n4) ((v4f*)p)[i] = (v4f){0.f, 0.f, 0.f, 0.f};
}

// ---------------------------------------------------------------------------
// Dense GEMM C[M,NCOL] = A[M,K] @ B[K,NCOL] using V_WMMA_F32_16X16X4_F32.
// One wave per 16x16 tile of C; NCOL/16 waves per block cover all columns;
// blockIdx.x walks M in 16-row tiles (M = 50000 = 3125 * 16, exact).
//
// f32 WMMA fragment layout (32-bit A 16x4): lanes 0-15 carry M rows, lane
// half selects K pair: VGPR0 = K = k+2*half, VGPR1 = K = k+2*half+1 -> one
// aligned float2 load per lane per k-step. B 4x16 mirrors this: lane half
// selects K pair, lane&15 selects N column.
// C/D layout: VGPR v, lane l -> row = 8*(l>>4) + v, col = l&15.
// ---------------------------------------------------------------------------
template <int K, int NCOL>
__global__ void gcn_gemm_wmma_f32(const float* __restrict__ A,
                                  const float* __restrict__ B,
                                  float* __restrict__ C) {
    const int wave  = threadIdx.x >> 5;      // n-tile within block
    const int lane  = threadIdx.x & 31;
    const int hi    = lane >> 4;             // lane half: 0 or 1
    const int l15   = lane & 15;
    const int mrow  = blockIdx.x * 16 + l15; // A row this lane feeds
    const int ncol  = wave * 16 + l15;       // B/C column this lane feeds

    v8f acc = {};
    #pragma unroll 4
    for (int k = 0; k < K; k += 4) {
        const int kk = k + 2 * hi;
        // A fragment: float2 at (mrow, kk) -- 8B aligned (kk even)
        v2f a = *(const v2f*)(A + (size_t)mrow * K + kk);
        // B fragment: rows kk, kk+1 at column ncol
        v2f b;
        b.x = B[(size_t)kk * NCOL + ncol];
        b.y = B[(size_t)(kk + 1) * NCOL + ncol];
        acc = __builtin_amdgcn_wmma_f32_16x16x4_f32(
            /*neg_a=*/false, a, /*neg_b=*/false, b,
            /*c_mod=*/(short)0, acc, /*reuse_a=*/false, /*reuse_b=*/false);
    }

    // Store D: lane writes 8 rows (stride NCOL) in column ncol.
    float* Cp = C + (size_t)(blockIdx.x * 16 + 8 * hi) * NCOL + ncol;
    #pragma unroll
    for (int v = 0; v < 8; ++v) Cp[(size_t)v * NCOL] = acc[v];
}

// ---------------------------------------------------------------------------
// Guaranteed-hardware f32 global atomic add (no return -> STOREcnt).
// gfx1250 ISA §15.18.3 op 86: GLOBAL_ATOMIC_ADD_F32, RNE rounding.
// s_endpgm performs an implicit wait-idle, so outstanding atomics drain.
// ---------------------------------------------------------------------------
__device__ __forceinline__ void global_fadd(float* p, float v) {
    asm volatile("global_atomic_add_f32 %0, %1, off"
                 :
                 : "v"((uint64_t)(uintptr_t)p), "v"(v)
                 : "memory");
}

// ---------------------------------------------------------------------------
// SpMM scatter: for each edge e, acc[dst] += S[src] * w[e].
// One wave per edge; lane handles PER_LANE consecutive features
// (aligned vector gather, hardware f32 global atomics for the scatter).
// edge_index is [2, E]: row 0 = dst, row 1 = src.
// ---------------------------------------------------------------------------
template <int F, int PER_LANE>
__global__ void gcn_spmm_atomic(const int* __restrict__ ei,
                                const float* __restrict__ ew,
                                const float* __restrict__ S,
                                float* __restrict__ acc, int E) {
    const int e    = (int)((blockIdx.x * (size_t)blockDim.x + threadIdx.x) >> 5);
    const int lane = threadIdx.x & 31;
    if (e >= E) return;
    const int dst  = ei[e];
    const int src  = ei[E + e];
    const float wt = ew[e];
    const float* sp = S + (size_t)src * F + lane * PER_LANE;
    float* ap       = acc + (size_t)dst * F + lane * PER_LANE;
    float vals[PER_LANE];
    #pragma unroll
    for (int j = 0; j < PER_LANE; ++j) vals[j] = sp[j];   // merges to b64/b128 load
    #pragma unroll
    for (int j = 0; j < PER_LANE; ++j)
        global_fadd(ap + j, vals[j] * wt);
}

// ---------------------------------------------------------------------------
// In-place bias add, optionally fused relu. F is a power of two.
// ---------------------------------------------------------------------------
template <int F, bool RELU>
__global__ void gcn_bias_act(float* __restrict__ x, const float* __restrict__ b,
                             int total) {
    int i = blockIdx.x * blockDim.x + threadIdx.x;
    if (i >= total) return;
    float v = x[i] + b[i & (F - 1)];
    if (RELU) v = fmaxf(v, 0.f);
    x[i] = v;
}

// ---------------------------------------------------------------------------
extern "C" void kernel_launch(void* const* d_in, const int* in_sizes, int n_in,
                              void* d_out, int out_size, void* d_ws, size_t ws_size,
                              hipStream_t stream) {
    const float* x  = (const float*)d_in[0];
    const int*   ei = (const int*)  d_in[1];
    const float* ew = (const float*)d_in[2];
    const float* W1 = (const float*)d_in[3];
    const float* b1 = (const float*)d_in[4];
    const float* W2 = (const float*)d_in[5];
    const float* b2 = (const float*)d_in[6];

    const int N = in_sizes[0] / 512;   // 50000
    const int E = in_sizes[2];         // 1600000

    float* out = (float*)d_out;
    float* x1  = out;                       // [N,128]
    float* x2  = out + (size_t)N * 128;     // [N,64]
    float* supp = (float*)d_ws;             // reused: [N,128] then [N,64]

    // 1) Zero both accumulator regions in d_out (N*192 floats, /4 exact).
    {
        int n4 = (N * 192) / 4;
        gcn_zero_f4<<<(n4 + 255) / 256, 256, 0, stream>>>(out, n4);
    }

    // 2) support1 = x @ W1   [N,512]x[512,128] -> ws
    gcn_gemm_wmma_f32<512, 128><<<N / 16, 256, 0, stream>>>(x, W1, supp);

    // 3) x1 += scatter(support1)   (hardware global_atomic_add_f32)
    gcn_spmm_atomic<128, 4><<<(E + 7) / 8, 256, 0, stream>>>(ei, ew, supp, x1, E);

    // 4) x1 = relu(x1 + b1) in place
    gcn_bias_act<128, true><<<(N * 128 + 255) / 256, 256, 0, stream>>>(x1, b1, N * 128);

    // 5) support2 = x1 @ W2   [N,128]x[128,64] -> ws (reuse)
    gcn_gemm_wmma_f32<128, 64><<<N / 16, 128, 0, stream>>>(x1, W2, supp);

    // 6) x2 += scatter(support2)
    gcn_spmm_atomic<64, 2><<<(E + 7) / 8, 256, 0, stream>>>(ei, ew, supp, x2, E);

    // 7) x2 = x2 + b2 in place
    gcn_bias_act<64, false><<<(N * 64 + 255) / 256, 256, 0, stream>>>(x2, b2, N * 64);
}